// PostProcess_36756330119453
// MI455X (gfx1250) — compile-verified
//
#include <hip/hip_runtime.h>
#include <math.h>

#define BN 8
#define HH 256
#define WW 256
#define HWSZ 65536
#define KK 100
#define JJ 17
#define NCH 144          // 8 hm channels + 136 hm_hp channels
#define CHUNK 2048
#define NBLK_PER_CH 32   // 65536 / 2048
#define NCAND (NBLK_PER_CH * KK)   // 3200
#define KPAD 112
#define THRESHV 0.1f
#define NEGBIG -3.4e38f

typedef float v2f __attribute__((ext_vector_type(2)));
typedef float v4f __attribute__((ext_vector_type(4)));
typedef float v8f __attribute__((ext_vector_type(8)));

// ---------------------------------------------------------------- streaming
__global__ void copy4_kernel(const v4f* __restrict__ s, v4f* __restrict__ d, int n4) {
    int i = blockIdx.x * blockDim.x + threadIdx.x;
    if (i < n4) {
        v4f v = __builtin_nontemporal_load(&s[i]);
        __builtin_nontemporal_store(v, &d[i]);
    }
}

__global__ void sigmoid4_kernel(const v4f* __restrict__ s, v4f* __restrict__ d, int n4) {
    int i = blockIdx.x * blockDim.x + threadIdx.x;
    if (i < n4) {
        v4f v = __builtin_nontemporal_load(&s[i]);
        v4f o;
        o.x = 1.0f / (1.0f + __expf(-v.x));
        o.y = 1.0f / (1.0f + __expf(-v.y));
        o.z = 1.0f / (1.0f + __expf(-v.z));
        o.w = 1.0f / (1.0f + __expf(-v.w));
        __builtin_nontemporal_store(o, &d[i]);
    }
}

// ------------------------------------------------- stage 1: NMS + local topK
// grid = NCH * NBLK_PER_CH blocks, 256 threads. Each block: 2048 elements of
// one channel -> local top-100 (score, flat idx) sorted descending.
__global__ void nms_topk_stage1(const float* __restrict__ hm,
                                const float* __restrict__ hmhp,
                                float* __restrict__ cand_s, int* __restrict__ cand_i) {
    int blk  = blockIdx.x;
    int ch   = blk / NBLK_PER_CH;
    int part = blk % NBLK_PER_CH;
    const float* base = (ch < BN) ? (hm + (size_t)ch * HWSZ)
                                  : (hmhp + (size_t)(ch - BN) * HWSZ);
    int off = part * CHUNK;
    int t   = threadIdx.x;

    float ls[8];
    int   li[8];
#pragma unroll
    for (int s = 0; s < 8; ++s) {
        int idx = off + s * 256 + t;
        int y = idx >> 8, x = idx & 255;
        float raw = base[idx];
        float m = raw;
        for (int dy = -1; dy <= 1; ++dy) {
            int yy = y + dy;
            if (yy < 0 || yy >= HH) continue;
            for (int dx = -1; dx <= 1; ++dx) {
                int xx = x + dx;
                if (xx < 0 || xx >= WW) continue;
                m = fmaxf(m, base[yy * WW + xx]);
            }
        }
        // sigmoid monotone: raw peak <=> sigmoid peak; non-peaks become 0
        ls[s] = (raw == m) ? (1.0f / (1.0f + __expf(-raw))) : 0.0f;
        li[s] = idx;
    }

    __shared__ float rs[256];
    __shared__ int   ri[256];
    __shared__ int   rslot[256];

    for (int r = 0; r < KK; ++r) {
        float bs = ls[0]; int bi = li[0]; int bsl = 0;
#pragma unroll
        for (int s = 1; s < 8; ++s) {
            if (ls[s] > bs || (ls[s] == bs && li[s] < bi)) { bs = ls[s]; bi = li[s]; bsl = s; }
        }
        rs[t] = bs; ri[t] = bi; rslot[t] = (t << 3) | bsl;
        __syncthreads();
        for (int w = 128; w > 0; w >>= 1) {
            if (t < w) {
                if (rs[t + w] > rs[t] || (rs[t + w] == rs[t] && ri[t + w] < ri[t])) {
                    rs[t] = rs[t + w]; ri[t] = ri[t + w]; rslot[t] = rslot[t + w];
                }
            }
            __syncthreads();
        }
        int win = rslot[0];
        if (t == 0) {
            cand_s[(size_t)ch * NCAND + part * KK + r] = rs[0];
            cand_i[(size_t)ch * NCAND + part * KK + r] = ri[0];
        }
        int wt = win >> 3, wsl = win & 7;
        if (wt == t) {
#pragma unroll
            for (int s2 = 0; s2 < 8; ++s2) if (s2 == wsl) ls[s2] = NEGBIG;
        }
        __syncthreads();
    }
}

// ------------------------------------------------- stage 2: merge to top-100
// grid = NCH blocks, 256 threads.
__global__ void topk_stage2(const float* __restrict__ cand_s, const int* __restrict__ cand_i,
                            float* __restrict__ tk_s, int* __restrict__ tk_i) {
    __shared__ float cs[NCAND];
    __shared__ int   ci[NCAND];
    __shared__ float rs[256];
    __shared__ int   ri[256];
    __shared__ int   rp[256];
    int ch = blockIdx.x, t = threadIdx.x;
    for (int i = t; i < NCAND; i += 256) {
        cs[i] = cand_s[(size_t)ch * NCAND + i];
        ci[i] = cand_i[(size_t)ch * NCAND + i];
    }
    __syncthreads();
    for (int r = 0; r < KK; ++r) {
        float bs = NEGBIG; int bi = 0x7fffffff; int bp = 0;
        for (int i = t; i < NCAND; i += 256) {
            if (cs[i] > bs || (cs[i] == bs && ci[i] < bi)) { bs = cs[i]; bi = ci[i]; bp = i; }
        }
        rs[t] = bs; ri[t] = bi; rp[t] = bp;
        __syncthreads();
        for (int w = 128; w > 0; w >>= 1) {
            if (t < w) {
                if (rs[t + w] > rs[t] || (rs[t + w] == rs[t] && ri[t + w] < ri[t])) {
                    rs[t] = rs[t + w]; ri[t] = ri[t + w]; rp[t] = rp[t + w];
                }
            }
            __syncthreads();
        }
        if (t == 0) {
            tk_s[ch * KK + r] = rs[0];
            tk_i[ch * KK + r] = ri[0];
            cs[rp[0]] = NEGBIG;
        }
        __syncthreads();
    }
}

// --------------------------------- stage 3: bbox/score/cls + regressed kps
// grid = BN, 128 threads (k < 100)
__global__ void det_prep(const float* __restrict__ wh, const float* __restrict__ reg,
                         const float* __restrict__ hps,
                         const float* __restrict__ tk_s, const int* __restrict__ tk_i,
                         float* __restrict__ det, float* __restrict__ bb_ws,
                         float* __restrict__ kps_ws) {
    int b = blockIdx.x, k = threadIdx.x;
    if (k >= KK) return;
    int   ind = tk_i[b * KK + k];
    float sc  = tk_s[b * KK + k];
    float xs = (float)(ind & 255);
    float ys = (float)(ind >> 8);
    float rx = reg[(size_t)(b * 2 + 0) * HWSZ + ind];
    float ry = reg[(size_t)(b * 2 + 1) * HWSZ + ind];
    float wx = wh [(size_t)(b * 2 + 0) * HWSZ + ind];
    float wy = wh [(size_t)(b * 2 + 1) * HWSZ + ind];
    float cx = xs + rx, cy = ys + ry;
    float l = cx - 0.5f * wx, tt = cy - 0.5f * wy;
    float rr = cx + 0.5f * wx, bb = cy + 0.5f * wy;
    float* drow = det + (size_t)(b * KK + k) * 40;
    drow[0] = l; drow[1] = tt; drow[2] = rr; drow[3] = bb;
    drow[4] = sc; drow[39] = 0.0f;  // c==1 -> cls 0
    float* bw = bb_ws + (size_t)(b * KK + k) * 4;
    bw[0] = l; bw[1] = tt; bw[2] = rr; bw[3] = bb;
    for (int j = 0; j < JJ; ++j) {
        float px = hps[(size_t)(b * (2 * JJ) + 2 * j    ) * HWSZ + ind] + xs;
        float py = hps[(size_t)(b * (2 * JJ) + 2 * j + 1) * HWSZ + ind] + ys;
        kps_ws[((size_t)(b * JJ + j) * KK + k) * 2 + 0] = px;
        kps_ws[((size_t)(b * JJ + j) * KK + k) * 2 + 1] = py;
    }
}

// ------------------------- stage 3b: candidate coords (hp_offset + masking)
// grid = BN*JJ, 128 threads (k < 100)
__global__ void hp_prep(const float* __restrict__ hpoff,
                        const float* __restrict__ tk_s, const int* __restrict__ tk_i,
                        float* __restrict__ hmc_ws, float* __restrict__ hms_ws) {
    int bj = blockIdx.x, k = threadIdx.x;
    if (k >= KK) return;
    int b  = bj / JJ;
    int ch = BN + bj;
    int   ind = tk_i[ch * KK + k];
    float s   = tk_s[ch * KK + k];
    float xs = (float)(ind % HH);
    float ys = (float)(ind / HH);
    float ox = hpoff[(size_t)(b * 2 + 0) * HWSZ + ind];
    float oy = hpoff[(size_t)(b * 2 + 1) * HWSZ + ind];
    float hx = xs + ox, hy = ys + oy;
    bool m = s > THRESHV;
    hms_ws[bj * KK + k]           = m ? s  : -1.0f;
    hmc_ws[(bj * KK + k) * 2 + 0] = m ? hx : -10000.0f;
    hmc_ws[(bj * KK + k) * 2 + 1] = m ? hy : -10000.0f;
}

// --------------- stage 4: WMMA distance matrix + argmin + keypoint gating
// grid = BN*JJ blocks, 32 threads (one wave). d2(k,k2) via rank-4 GEMM:
//   A row k  = [kx, ky, 1, 0]                  (16x4 f32)
//   B col k2 = [-2cx, -2cy, cx^2+cy^2, 0]      (4x16 f32)
//   d2 = (kx^2+ky^2) + (A x B)[k][k2]
__global__ void __launch_bounds__(32)
kp_assign(const float* __restrict__ kps_ws, const float* __restrict__ hmc_ws,
          const float* __restrict__ hms_ws, const float* __restrict__ bb_ws,
          float* __restrict__ det) {
    __shared__ float kxl[KPAD], kyl[KPAD], cxl[KPAD], cyl[KPAD];
    __shared__ float mind[KPAD];
    __shared__ int   marg[KPAD];

    int bj = blockIdx.x;
    int b  = bj / JJ;
    int j  = bj % JJ;
    int lane = threadIdx.x;
    int half = lane >> 4;     // 0: K rows 0..1 of A / 1: K rows 2..3
    int l16  = lane & 15;

    for (int i = lane; i < KPAD; i += 32) {
        if (i < KK) {
            kxl[i] = kps_ws[((size_t)bj * KK + i) * 2 + 0];
            kyl[i] = kps_ws[((size_t)bj * KK + i) * 2 + 1];
            cxl[i] = hmc_ws[((size_t)bj * KK + i) * 2 + 0];
            cyl[i] = hmc_ws[((size_t)bj * KK + i) * 2 + 1];
        } else {
            kxl[i] = 0.0f; kyl[i] = 0.0f;
            cxl[i] = 1.0e8f; cyl[i] = 1.0e8f;   // pad cols -> huge distance
        }
    }
    __syncthreads();

    for (int rt = 0; rt < 7; ++rt) {
        int arow = rt * 16 + l16;
        v2f a;
        a.x = half ? 1.0f : kxl[arow];   // lanes 0-15: K=0 (kx), lanes 16-31: K=2 (1)
        a.y = half ? 0.0f : kyl[arow];   // lanes 0-15: K=1 (ky), lanes 16-31: K=3 (0)

        float minv[8]; int minc[8];
#pragma unroll
        for (int i = 0; i < 8; ++i) { minv[i] = 3.4e38f; minc[i] = 0; }

        for (int ct = 0; ct < 7; ++ct) {
            int col = ct * 16 + l16;
            float cx = cxl[col], cy = cyl[col];
            v2f bf;
            bf.x = half ? (cx * cx + cy * cy) : (-2.0f * cx);  // K=2 vs K=0
            bf.y = half ? 0.0f : (-2.0f * cy);                 // K=3 vs K=1
            v8f c = {0.f, 0.f, 0.f, 0.f, 0.f, 0.f, 0.f, 0.f};
            c = __builtin_amdgcn_wmma_f32_16x16x4_f32(
                    /*neg_a=*/false, a, /*neg_b=*/false, bf,
                    /*c_mod=*/(short)0, c, /*reuse_a=*/false, /*reuse_b=*/false);
#pragma unroll
            for (int i = 0; i < 8; ++i) {
                int row = rt * 16 + half * 8 + i;   // C layout: VGPR i -> M=i / M=i+8
                float rn = kxl[row] * kxl[row] + kyl[row] * kyl[row];
                float d2 = c[i] + rn;
                if (d2 < minv[i]) { minv[i] = d2; minc[i] = col; }
            }
        }
        // min/argmin across the 16 lanes of each half (cols), tie -> lowest col
        for (int off = 1; off < 16; off <<= 1) {
#pragma unroll
            for (int i = 0; i < 8; ++i) {
                float ov = __shfl_xor(minv[i], off, 32);
                int   oc = __shfl_xor(minc[i], off, 32);
                if (ov < minv[i] || (ov == minv[i] && oc < minc[i])) {
                    minv[i] = ov; minc[i] = oc;
                }
            }
        }
        if (l16 == 0) {
#pragma unroll
            for (int i = 0; i < 8; ++i) {
                int row = rt * 16 + half * 8 + i;
                mind[row] = minv[i];
                marg[row] = minc[i];
            }
        }
        __syncthreads();
    }

    for (int k = lane; k < KK; k += 32) {
        float d2 = fmaxf(mind[k], 0.0f);
        float md = sqrtf(d2);
        int   mi = marg[k];
        float hx = cxl[mi], hy = cyl[mi];
        float hs = (mi < KK) ? hms_ws[bj * KK + mi] : -1.0f;
        const float* bw = bb_ws + (size_t)(b * KK + k) * 4;
        float l = bw[0], t = bw[1], r = bw[2], bo = bw[3];
        bool bad = (hx < l) || (hx > r) || (hy < t) || (hy > bo) ||
                   (hs < THRESHV) || (md > fmaxf(bo - t, r - l) * 0.3f);
        float ox = bad ? kxl[k] : hx;
        float oy = bad ? kyl[k] : hy;
        float* drow = det + (size_t)(b * KK + k) * 40;
        drow[5 + 2 * j + 0] = ox;
        drow[5 + 2 * j + 1] = oy;
    }
}

// ---------------------------------------------------------------- launcher
extern "C" void kernel_launch(void* const* d_in, const int* in_sizes, int n_in,
                              void* d_out, int out_size, void* d_ws, size_t ws_size,
                              hipStream_t stream) {
    const float* hm    = (const float*)d_in[0];  // B*1*HW
    const float* wh    = (const float*)d_in[1];  // B*2*HW
    const float* hps   = (const float*)d_in[2];  // B*34*HW
    const float* reg   = (const float*)d_in[3];  // B*2*HW
    const float* hmhp  = (const float*)d_in[4];  // B*17*HW
    const float* hpoff = (const float*)d_in[5];  // B*2*HW

    const int n_hm    = BN * 1  * HWSZ;   // 524288
    const int n_wh    = BN * 2  * HWSZ;   // 1048576
    const int n_hps   = BN * 34 * HWSZ;   // 17825792
    const int n_reg   = n_wh;
    const int n_hmhp  = BN * 17 * HWSZ;   // 8912896
    const int n_hpoff = n_wh;

    float* out     = (float*)d_out;
    float* o_hm    = out;
    float* o_wh    = o_hm    + n_hm;
    float* o_hps   = o_wh    + n_wh;
    float* o_reg   = o_hps   + n_hps;
    float* o_hmhp  = o_reg   + n_reg;
    float* o_hpoff = o_hmhp  + n_hmhp;
    float* o_det   = o_hpoff + n_hpoff;   // B*K*40

    char* w = (char*)d_ws;
    float* cand_s = (float*)w; w += (size_t)NCH * NCAND * 4;
    int*   cand_i = (int*)  w; w += (size_t)NCH * NCAND * 4;
    float* tk_s   = (float*)w; w += (size_t)NCH * KK * 4;
    int*   tk_i   = (int*)  w; w += (size_t)NCH * KK * 4;
    float* bb_ws  = (float*)w; w += (size_t)BN * KK * 4 * 4;
    float* kps_ws = (float*)w; w += (size_t)BN * JJ * KK * 2 * 4;
    float* hmc_ws = (float*)w; w += (size_t)BN * JJ * KK * 2 * 4;
    float* hms_ws = (float*)w; w += (size_t)BN * JJ * KK * 4;

    // bandwidth-bound pass-throughs / sigmoids (float4, non-temporal)
    auto g4 = [](int n) { return ((n / 4) + 255) / 256; };
    sigmoid4_kernel<<<g4(n_hm),    256, 0, stream>>>((const v4f*)hm,    (v4f*)o_hm,    n_hm    / 4);
    copy4_kernel   <<<g4(n_wh),    256, 0, stream>>>((const v4f*)wh,    (v4f*)o_wh,    n_wh    / 4);
    copy4_kernel   <<<g4(n_hps),   256, 0, stream>>>((const v4f*)hps,   (v4f*)o_hps,   n_hps   / 4);
    copy4_kernel   <<<g4(n_reg),   256, 0, stream>>>((const v4f*)reg,   (v4f*)o_reg,   n_reg   / 4);
    sigmoid4_kernel<<<g4(n_hmhp),  256, 0, stream>>>((const v4f*)hmhp,  (v4f*)o_hmhp,  n_hmhp  / 4);
    copy4_kernel   <<<g4(n_hpoff), 256, 0, stream>>>((const v4f*)hpoff, (v4f*)o_hpoff, n_hpoff / 4);

    // detection pipeline
    nms_topk_stage1<<<NCH * NBLK_PER_CH, 256, 0, stream>>>(hm, hmhp, cand_s, cand_i);
    topk_stage2    <<<NCH,               256, 0, stream>>>(cand_s, cand_i, tk_s, tk_i);
    det_prep       <<<BN,                128, 0, stream>>>(wh, reg, hps, tk_s, tk_i, o_det, bb_ws, kps_ws);
    hp_prep        <<<BN * JJ,           128, 0, stream>>>(hpoff, tk_s, tk_i, hmc_ws, hms_ws);
    kp_assign      <<<BN * JJ,            32, 0, stream>>>(kps_ws, hmc_ws, hms_ws, bb_ws, o_det);
}